// BlurPool_83769042141665
// MI455X (gfx1250) — compile-verified
//
#include <hip/hip_runtime.h>

// BlurPool 5x5x5 binomial, replicate pad 2, stride 2, fused separable.
// x: [2,32,96,96,96] f32  ->  y: [2,32,48,48,48] f32
//
// Bandwidth-bound (255 MB @ 23.3 TB/s ~= 11 us floor). Uses the CDNA5 async
// global->LDS data mover (global_load_async_to_lds_b128 + ASYNCcnt) with
// double buffering; WMMA intentionally unused (banded f32 matmul would waste
// MACs and this kernel is nowhere near the VALU ceiling anyway).

constexpr int S        = 96;          // input spatial extent
constexpr int O        = 48;          // output spatial extent
constexpr int PLANE_IN = S * S;       // 9216 floats = 36 KB
constexpr int TMP_ELEMS= S * O;       // 4608 floats = 18 KB
constexpr int OUT_PLANE= O * O;       // 2304 floats
constexpr int NTHREADS = 256;         // 8 wave32s

__device__ __forceinline__ int clampi(int v, int lo, int hi) {
    return v < lo ? lo : (v > hi ? hi : v);
}

__global__ __launch_bounds__(NTHREADS) void blurpool3d_kernel(
        const float* __restrict__ x, float* __restrict__ y) {
    const int zo  = blockIdx.x;   // 0..47 output depth
    const int nc  = blockIdx.y;   // 0..63 fused N*C
    const int tid = threadIdx.x;

    extern __shared__ float smem[];
    float* bufs = smem;                  // 2 * 9216 floats (double buffer)
    float* tmp  = smem + 2 * PLANE_IN;   // 4608 floats (W-blurred intermediate)

    const float* img = x + (size_t)nc * (size_t)(S * S * S);

    // binomial taps /16
    constexpr float F0 = 1.0f / 16.0f;
    constexpr float F1 = 4.0f / 16.0f;
    constexpr float F2 = 6.0f / 16.0f;

    float acc[9];
#pragma unroll
    for (int k = 0; k < 9; ++k) acc[k] = 0.0f;

    // Issue async copy of input D-plane for tap kd into LDS buffer (kd&1).
    auto issue_plane = [&](int kd) {
        const int d = clampi(2 * zo + kd - 2, 0, S - 1);   // replicate pad in D
        const float*  p   = img + (size_t)d * PLANE_IN;
        float*        dst = bufs + (kd & 1) * PLANE_IN;
#pragma unroll
        for (int i = 0; i < 9; ++i) {                       // 9*256 = 2304 float4
            const int e = tid + i * NTHREADS;               // float4 index
            uint32_t lds_addr = (uint32_t)(uintptr_t)(dst + 4 * e);
            const float4* g   = (const float4*)p + e;
            asm volatile("global_load_async_to_lds_b128 %0, %1, off"
                         :: "v"(lds_addr), "v"(g) : "memory");
        }
    };

    issue_plane(0);

    for (int kd = 0; kd < 5; ++kd) {
        if (kd < 4) {
            issue_plane(kd + 1);                     // prefetch next plane
            // 18 async ops in flight per wave; <=9 left => plane kd complete
            // (async loads complete in issue order)
            asm volatile("s_wait_asynccnt 0x9" ::: "memory");
        } else {
            asm volatile("s_wait_asynccnt 0x0" ::: "memory");
        }
        __syncthreads();                             // plane kd visible to all waves

        const float* in = bufs + (kd & 1) * PLANE_IN;

        // ---- W blur (stride 2, replicate clamp): tmp[h][wo], 4608 elems ----
#pragma unroll
        for (int i = 0; i < 18; ++i) {
            const int e  = tid + i * NTHREADS;       // 0..4607
            const int h  = e / O;
            const int wo = e - h * O;
            const int w0 = 2 * wo - 2;
            const float* row = in + h * S;
            float s;
            s  = F0 * row[clampi(w0 + 0, 0, S - 1)];
            s += F1 * row[clampi(w0 + 1, 0, S - 1)];
            s += F2 * row[clampi(w0 + 2, 0, S - 1)];
            s += F1 * row[clampi(w0 + 3, 0, S - 1)];
            s += F0 * row[clampi(w0 + 4, 0, S - 1)];
            tmp[e] = s;
        }
        __syncthreads();                             // tmp ready; buf reads done

        // ---- H blur (stride 2) + D-tap weight, accumulate in registers ----
        const float fd = (kd == 0 || kd == 4) ? F0 : ((kd == 1 || kd == 3) ? F1 : F2);
#pragma unroll
        for (int k = 0; k < 9; ++k) {
            const int e  = tid + k * NTHREADS;       // 0..2303
            const int ho = e / O;
            const int wo = e - ho * O;
            const int h0 = 2 * ho - 2;
            float s;
            s  = F0 * tmp[clampi(h0 + 0, 0, S - 1) * O + wo];
            s += F1 * tmp[clampi(h0 + 1, 0, S - 1) * O + wo];
            s += F2 * tmp[clampi(h0 + 2, 0, S - 1) * O + wo];
            s += F1 * tmp[clampi(h0 + 3, 0, S - 1) * O + wo];
            s += F0 * tmp[clampi(h0 + 4, 0, S - 1) * O + wo];
            acc[k] += fd * s;
        }
        // next iteration's issue_plane overwrites buf[(kd+1)&1]; its last readers
        // (stage-W of iteration kd-1) are fenced by this iteration's barriers.
    }

    // ---- coalesced non-temporal store of the 48x48 output plane ----
    float* outp = y + (size_t)(nc * O + zo) * OUT_PLANE;
#pragma unroll
    for (int k = 0; k < 9; ++k) {
        const int e = tid + k * NTHREADS;
        __builtin_nontemporal_store(acc[k], outp + e);
    }
}

extern "C" void kernel_launch(void* const* d_in, const int* in_sizes, int n_in,
                              void* d_out, int out_size, void* d_ws, size_t ws_size,
                              hipStream_t stream) {
    (void)in_sizes; (void)n_in; (void)out_size; (void)d_ws; (void)ws_size;
    const float* x = (const float*)d_in[0];
    float*       y = (float*)d_out;

    const size_t shmem = (size_t)(2 * PLANE_IN + TMP_ELEMS) * sizeof(float); // 92160 B
    hipFuncSetAttribute((const void*)blurpool3d_kernel,
                        hipFuncAttributeMaxDynamicSharedMemorySize, (int)shmem);

    dim3 grid(O /*zo=48*/, 64 /*n*c*/);
    blurpool3d_kernel<<<grid, NTHREADS, shmem, stream>>>(x, y);
}